// GCNVANet_2576980378008
// MI455X (gfx1250) — compile-verified
//
#include <hip/hip_runtime.h>
#include <math.h>

typedef __attribute__((ext_vector_type(2))) float v2f;
typedef __attribute__((ext_vector_type(8))) float v8f;

#define DD 20
#define ACTN 5
#define NEG_SLOPE 0.2f

// ---------- helpers: ordered-uint encoding for float atomicMax ----------
__device__ __forceinline__ unsigned f2o(float f) {
  unsigned u = __float_as_uint(f);
  return (u & 0x80000000u) ? ~u : (u | 0x80000000u);
}
__device__ __forceinline__ float o2f(unsigned u) {
  return (u & 0x80000000u) ? __uint_as_float(u & 0x7fffffffu)
                           : __uint_as_float(~u);
}

// ---------- fills ----------
__global__ void fill_f32(float* p, float v, int n) {
  int i = blockIdx.x * blockDim.x + threadIdx.x;
  if (i < n) p[i] = v;
}
__global__ void fill_u32(unsigned* p, unsigned v, int n) {
  int i = blockIdx.x * blockDim.x + threadIdx.x;
  if (i < n) p[i] = v;
}

// ---------- degree (in-degree over dst, incl. self loops) ----------
__global__ void deg_kernel(const int* __restrict__ dstI, int E, int ET,
                           float* __restrict__ deg) {
  int e = blockIdx.x * blockDim.x + threadIdx.x;
  if (e >= ET) return;
  int d = (e < E) ? dstI[e] : (e - E);
  atomicAdd(deg + d, 1.0f);
}
__global__ void rsqrt_kernel(float* p, int n) {
  int i = blockIdx.x * blockDim.x + threadIdx.x;
  if (i < n) p[i] = rsqrtf(p[i]);   // deg >= 1 always (self loops)
}

// ---------- WMMA fp32 GEMM: Y[n x 20] = X[n x 20] @ W[20 x 20] ----------
// W staged once per block into LDS, zero-padded to 20x32 -> all B fragment
// reads are unconditional ds_loads (no EXEC divergence around the WMMAs).
// One wave32 per 16-row tile; K=20 -> 5 chained V_WMMA_F32_16X16X4_F32 per
// column tile (cols 0-15 and 16-31).
__global__ void gemm20_wmma(const float* __restrict__ X, const float* __restrict__ W,
                            float* __restrict__ Y, int n) {
  __shared__ float WS[DD][32];
  for (int i = threadIdx.x; i < DD * 32; i += blockDim.x) {
    int r = i >> 5, c = i & 31;
    WS[r][c] = (c < DD) ? W[r * DD + c] : 0.0f;
  }
  __syncthreads();

  int wavesPerBlock = blockDim.x >> 5;
  int wave = blockIdx.x * wavesPerBlock + (threadIdx.x >> 5);
  int nTiles = (n + 15) >> 4;
  if (wave >= nTiles) return;           // uniform per-wave exit
  int lane = threadIdx.x & 31;
  int half = lane >> 4;                 // 0: K pair {4k,4k+1}, 1: {4k+2,4k+3}
  int l = lane & 15;
  int row0 = wave << 4;
  int rowA = row0 + l;
  if (rowA >= n) rowA = n - 1;          // branchless clamp (stores guarded)
  const float* xr = X + (size_t)rowA * DD;

  v8f acc0 = {0.f,0.f,0.f,0.f,0.f,0.f,0.f,0.f};
  v8f acc1 = {0.f,0.f,0.f,0.f,0.f,0.f,0.f,0.f};
#pragma unroll
  for (int k = 0; k < 5; ++k) {
    int kb = 4 * k + 2 * half;
    v2f a;  a.x  = xr[kb];          a.y  = xr[kb + 1];
    v2f b0; b0.x = WS[kb][l];       b0.y = WS[kb + 1][l];
    acc0 = __builtin_amdgcn_wmma_f32_16x16x4_f32(false, a, false, b0,
                                                 (short)0, acc0, false, false);
    v2f b1; b1.x = WS[kb][16 + l];  b1.y = WS[kb + 1][16 + l];
    acc1 = __builtin_amdgcn_wmma_f32_16x16x4_f32(false, a, false, b1,
                                                 (short)0, acc1, false, false);
  }
  // D layout: acc[r] -> (M = r + half*8, N = l) / (N = 16+l) for tile 1
  int mBase = row0 + half * 8;
#pragma unroll
  for (int r = 0; r < 8; ++r) {
    int m = mBase + r;
    if (m < n) {
      Y[(size_t)m * DD + l] = acc0[r];
      if (l < DD - 16) Y[(size_t)m * DD + 16 + l] = acc1[r];
    }
  }
}

// ---------- GCN message scatter: agg[dst] += dinv[s]*dinv[d] * h[src] ----------
__global__ void gcn_scatter(const int* __restrict__ srcI, const int* __restrict__ dstI,
                            int E, int ET, const float* __restrict__ dinv,
                            const float* __restrict__ h, float* __restrict__ agg) {
  int e = blockIdx.x * blockDim.x + threadIdx.x;
  if (e >= ET) return;
  int s, d;
  if (e < E) { s = srcI[e]; d = dstI[e]; } else { s = e - E; d = s; }
  float norm = dinv[s] * dinv[d];
  const float4* hr = (const float4*)(h + (size_t)s * DD);  // row = 80B, 16B aligned
  float* ag = agg + (size_t)d * DD;
#pragma unroll
  for (int q = 0; q < 5; ++q) {
    float4 v = hr[q];
    atomicAdd(ag + 4 * q + 0, norm * v.x);
    atomicAdd(ag + 4 * q + 1, norm * v.y);
    atomicAdd(ag + 4 * q + 2, norm * v.z);
    atomicAdd(ag + 4 * q + 3, norm * v.w);
  }
}

// ---------- bias + relu, in place ----------
__global__ void bias_relu(float* __restrict__ h, const float* __restrict__ b, int total) {
  int i = blockIdx.x * blockDim.x + threadIdx.x;
  if (i >= total) return;
  float v = h[i] + b[i % DD];
  h[i] = fmaxf(v, 0.0f);
}

// ---------- per-node attention dots: a_s = hg@att_src, a_d = hg@att_dst ----------
__global__ void as_ad_kernel(const float* __restrict__ hg, const float* __restrict__ attS,
                             const float* __restrict__ attD, float* __restrict__ asv,
                             float* __restrict__ adv, int n) {
  int i = blockIdx.x * blockDim.x + threadIdx.x;
  if (i >= n) return;
  const float* r = hg + (size_t)i * DD;
  float sa = 0.f, sd = 0.f;
#pragma unroll
  for (int k = 0; k < DD; ++k) { float v = r[k]; sa += v * attS[k]; sd += v * attD[k]; }
  asv[i] = sa; adv[i] = sd;
}

// ---------- GAT edge pass 1: e = leaky_relu(a_s[s]+a_d[d]); segment max ----------
__global__ void gat_edge_max(const int* __restrict__ srcI, const int* __restrict__ dstI,
                             int E, int ET, const float* __restrict__ asv,
                             const float* __restrict__ adv, float* __restrict__ ev,
                             unsigned* __restrict__ mmax) {
  int e = blockIdx.x * blockDim.x + threadIdx.x;
  if (e >= ET) return;
  int s, d;
  if (e < E) { s = srcI[e]; d = dstI[e]; } else { s = e - E; d = s; }
  float x = asv[s] + adv[d];
  x = (x > 0.0f) ? x : NEG_SLOPE * x;
  ev[e] = x;
  atomicMax(mmax + d, f2o(x));
}

// ---------- GAT edge pass 2: ex = exp(e - m[d]); denom scatter ----------
__global__ void gat_edge_exp(const int* __restrict__ srcI, const int* __restrict__ dstI,
                             int E, int ET, float* __restrict__ ev,
                             const unsigned* __restrict__ mmax, float* __restrict__ den) {
  int e = blockIdx.x * blockDim.x + threadIdx.x;
  if (e >= ET) return;
  int d = (e < E) ? dstI[e] : (e - E);
  float ex = __expf(ev[e] - o2f(mmax[d]));
  ev[e] = ex;
  atomicAdd(den + d, ex);
}

// ---------- GAT edge pass 3: agg[d] += (ex/denom[d]) * hg[s] ----------
__global__ void gat_scatter(const int* __restrict__ srcI, const int* __restrict__ dstI,
                            int E, int ET, const float* __restrict__ ev,
                            const float* __restrict__ den, const float* __restrict__ hg,
                            float* __restrict__ agg) {
  int e = blockIdx.x * blockDim.x + threadIdx.x;
  if (e >= ET) return;
  int s, d;
  if (e < E) { s = srcI[e]; d = dstI[e]; } else { s = e - E; d = s; }
  float alpha = ev[e] / den[d];
  const float4* hr = (const float4*)(hg + (size_t)s * DD);
  float* ag = agg + (size_t)d * DD;
#pragma unroll
  for (int q = 0; q < 5; ++q) {
    float4 v = hr[q];
    atomicAdd(ag + 4 * q + 0, alpha * v.x);
    atomicAdd(ag + 4 * q + 1, alpha * v.y);
    atomicAdd(ag + 4 * q + 2, alpha * v.z);
    atomicAdd(ag + 4 * q + 3, alpha * v.w);
  }
}

// ---------- graph readout: g[col] = sum_n h[n][col]; one block per column ----------
__global__ void readout_kernel(const float* __restrict__ h, float* __restrict__ g, int n) {
  int col = blockIdx.x;  // 0..19
  float s = 0.0f;
  for (int i = threadIdx.x; i < n; i += blockDim.x) s += h[(size_t)i * DD + col];
  __shared__ float red[256];
  red[threadIdx.x] = s;
  __syncthreads();
  for (int off = blockDim.x >> 1; off > 0; off >>= 1) {
    if ((int)threadIdx.x < off) red[threadIdx.x] += red[threadIdx.x + off];
    __syncthreads();
  }
  if (threadIdx.x == 0) g[col] = red[0];
}

// ---------- dueling head: single wave ----------
__global__ void head_kernel(const float* __restrict__ g,
                            const float* __restrict__ aW1, const float* __restrict__ ab1,
                            const float* __restrict__ aW2, const float* __restrict__ ab2,
                            const float* __restrict__ vW1, const float* __restrict__ vb1,
                            const float* __restrict__ vW2, const float* __restrict__ vb2,
                            float* __restrict__ out) {
  __shared__ float hA[DD], hV[DD], Ab[ACTN], Vv;
  int t = threadIdx.x;
  if (t < DD) {
    float sa = ab1[t], sv = vb1[t];
#pragma unroll
    for (int k = 0; k < DD; ++k) {
      float gk = g[k];
      sa += gk * aW1[k * DD + t];
      sv += gk * vW1[k * DD + t];
    }
    hA[t] = fmaxf(sa, 0.0f);
    hV[t] = fmaxf(sv, 0.0f);
  }
  __syncthreads();
  if (t < ACTN) {
    float s = ab2[t];
#pragma unroll
    for (int j = 0; j < DD; ++j) s += hA[j] * aW2[j * ACTN + t];
    Ab[t] = s;
  }
  if (t == ACTN) {
    float s = vb2[0];
#pragma unroll
    for (int j = 0; j < DD; ++j) s += hV[j] * vW2[j];
    Vv = s;
  }
  __syncthreads();
  if (t == 0) {
    float mean = 0.0f;
#pragma unroll
    for (int a = 0; a < ACTN; ++a) mean += Ab[a];
    mean *= (1.0f / ACTN);
#pragma unroll
    for (int a = 0; a < ACTN; ++a) out[a] = Vv + Ab[a] - mean;
  }
}

static inline int gridFor(int n, int b) { return (n + b - 1) / b; }

extern "C" void kernel_launch(void* const* d_in, const int* in_sizes, int n_in,
                              void* d_out, int out_size, void* d_ws, size_t ws_size,
                              hipStream_t stream) {
  const float* x    = (const float*)d_in[0];
  const int*   ei   = (const int*)d_in[1];
  // d_in[2] = edge_attr (unused by reference)
  const float* gcnW = (const float*)d_in[3];
  const float* gcnB = (const float*)d_in[4];
  const float* gatW = (const float*)d_in[5];
  const float* attS = (const float*)d_in[6];
  const float* attD = (const float*)d_in[7];
  const float* gatB = (const float*)d_in[8];
  const float* aW1  = (const float*)d_in[9];
  const float* ab1  = (const float*)d_in[10];
  const float* aW2  = (const float*)d_in[11];
  const float* ab2  = (const float*)d_in[12];
  const float* vW1  = (const float*)d_in[13];
  const float* vb1  = (const float*)d_in[14];
  const float* vW2  = (const float*)d_in[15];
  const float* vb2  = (const float*)d_in[16];

  const int N  = in_sizes[0] / DD;
  const int E  = in_sizes[1] / 2;
  const int ET = E + N;
  const int* srcI = ei;
  const int* dstI = ei + E;

  // ---- scratch layout (256B aligned chunks) ----
  char* wsp = (char*)d_ws;
  size_t off = 0;
  auto alloc = [&](size_t bytes) -> char* {
    char* p = wsp + off;
    off = (off + bytes + 255) & ~(size_t)255;
    return p;
  };
  float*    h1   = (float*)alloc(sizeof(float) * (size_t)N * DD);  // GEMM output (h / hg)
  float*    agg  = (float*)alloc(sizeof(float) * (size_t)N * DD);  // aggregation buffer
  float*    dinv = (float*)alloc(sizeof(float) * (size_t)N);
  float*    asv  = (float*)alloc(sizeof(float) * (size_t)N);
  float*    adv  = (float*)alloc(sizeof(float) * (size_t)N);
  unsigned* mmax = (unsigned*)alloc(sizeof(unsigned) * (size_t)N);
  float*    den  = (float*)alloc(sizeof(float) * (size_t)N);
  float*    ev   = (float*)alloc(sizeof(float) * (size_t)ET);
  float*    g    = (float*)alloc(sizeof(float) * 32);

  const int B = 256;
  const int nTiles = (N + 15) >> 4;
  const int gemmBlocks = gridFor(nTiles, B / 32);

  // ---- 1. degrees & symmetric norm ----
  fill_f32<<<gridFor(N, B), B, 0, stream>>>(dinv, 0.0f, N);
  deg_kernel<<<gridFor(ET, B), B, 0, stream>>>(dstI, E, ET, dinv);
  rsqrt_kernel<<<gridFor(N, B), B, 0, stream>>>(dinv, N);

  // ---- 2. GCN: h1 = x @ gcn_W (WMMA) ----
  gemm20_wmma<<<gemmBlocks, B, 0, stream>>>(x, gcnW, h1, N);

  // ---- 3. GCN scatter, bias, relu ----
  fill_f32<<<gridFor(N * DD, B), B, 0, stream>>>(agg, 0.0f, N * DD);
  gcn_scatter<<<gridFor(ET, B), B, 0, stream>>>(srcI, dstI, E, ET, dinv, h1, agg);
  bias_relu<<<gridFor(N * DD, B), B, 0, stream>>>(agg, gcnB, N * DD);

  // ---- 4. GAT: hg = relu_out @ gat_W (WMMA, reuse h1 buffer) ----
  gemm20_wmma<<<gemmBlocks, B, 0, stream>>>(agg, gatW, h1, N);
  as_ad_kernel<<<gridFor(N, B), B, 0, stream>>>(h1, attS, attD, asv, adv, N);

  // ---- 5. attention softmax over incoming edges ----
  fill_u32<<<gridFor(N, B), B, 0, stream>>>(mmax, 0x007FFFFFu, N);  // f2o(-inf)
  fill_f32<<<gridFor(N, B), B, 0, stream>>>(den, 0.0f, N);
  gat_edge_max<<<gridFor(ET, B), B, 0, stream>>>(srcI, dstI, E, ET, asv, adv, ev, mmax);
  gat_edge_exp<<<gridFor(ET, B), B, 0, stream>>>(srcI, dstI, E, ET, ev, mmax, den);

  // ---- 6. GAT weighted scatter, bias, relu ----
  fill_f32<<<gridFor(N * DD, B), B, 0, stream>>>(agg, 0.0f, N * DD);
  gat_scatter<<<gridFor(ET, B), B, 0, stream>>>(srcI, dstI, E, ET, ev, den, h1, agg);
  bias_relu<<<gridFor(N * DD, B), B, 0, stream>>>(agg, gatB, N * DD);

  // ---- 7. readout + dueling head ----
  readout_kernel<<<DD, 256, 0, stream>>>(agg, g, N);
  head_kernel<<<1, 32, 0, stream>>>(g, aW1, ab1, aW2, ab2, vW1, vb1, vW2, vb2,
                                    (float*)d_out);
}